// AttnDecoderRNN_36773509988584
// MI455X (gfx1250) — compile-verified
//
#include <hip/hip_runtime.h>
#include <math.h>

#define H 1024
#define V 50257
#define S 8192

typedef float v2f __attribute__((ext_vector_type(2)));
typedef float v8f __attribute__((ext_vector_type(8)));

// ---- workspace layout (float offsets) ----
#define WS_LOGITS 0                      // V floats (padded to 50432)
#define WS_CAT    50432                  // 2H
#define WS_X      (WS_CAT + 2*H)         // H
#define WS_GATES  (WS_X + H)             // 6H  (gi[3H] then gh[3H])
#define WS_HN     (WS_GATES + 6*H)       // H
#define WS_PART   (WS_HN + H)            // 64*H partial column sums

// attn_weights == softmax(constant) == exactly 1/S
__global__ void k_fill_attn(float* __restrict__ out_attn) {
  int i = blockIdx.x * blockDim.x + threadIdx.x;
  if (i < S) out_attn[i] = 1.0f / (float)S;
}

// Partial column sums of encoder_outputs: 64 blocks x 128 rows, coalesced.
__global__ void k_colsum_partial(const float* __restrict__ enc, float* __restrict__ part) {
  int t = threadIdx.x;          // 0..255
  int b = blockIdx.x;           // 0..63
  float a0 = 0.f, a1 = 0.f, a2 = 0.f, a3 = 0.f;
  const float* p = enc + (size_t)b * 128 * H;
  for (int r = 0; r < 128; ++r) {
    const float* row = p + (size_t)r * H;
    a0 += row[t]; a1 += row[t + 256]; a2 += row[t + 512]; a3 += row[t + 768];
  }
  float* dst = part + b * H;
  dst[t] = a0; dst[t + 256] = a1; dst[t + 512] = a2; dst[t + 768] = a3;
}

// Reduce partials -> attn_applied, and gather embedding row -> cat = [emb, aa]
__global__ void k_build_cat(const float* __restrict__ part, const int* __restrict__ tok,
                            const float* __restrict__ emb_tbl, float* __restrict__ cat) {
  int j = blockIdx.x * blockDim.x + threadIdx.x;   // 0..1023
  float s = 0.f;
  for (int k = 0; k < 64; ++k) s += part[k * H + j];
  cat[H + j] = s * (1.0f / (float)S);
  int token = tok[0];
  cat[j] = emb_tbl[(size_t)token * H + j];
}

// combined[row] = relu( dot(comb_w[row, 0:2H], cat) + comb_b[row] )
__global__ void k_combined(const float* __restrict__ comb_w, const float* __restrict__ comb_b,
                           const float* __restrict__ cat, float* __restrict__ x) {
  __shared__ float red[256];
  int row = blockIdx.x, t = threadIdx.x;
  const float* wr = comb_w + (size_t)row * (2 * H);
  float acc = 0.f;
  for (int i = t; i < 2 * H; i += 256) acc += wr[i] * cat[i];
  red[t] = acc; __syncthreads();
  for (int s2 = 128; s2 > 0; s2 >>= 1) { if (t < s2) red[t] += red[t + s2]; __syncthreads(); }
  if (t == 0) { float v = red[0] + comb_b[row]; x[row] = v > 0.f ? v : 0.f; }
}

// gates[0:3H] = w_ih@x + b_ih ; gates[3H:6H] = w_hh@h0 + b_hh   (block per output row)
__global__ void k_gates(const float* __restrict__ w_ih, const float* __restrict__ w_hh,
                        const float* __restrict__ b_ih, const float* __restrict__ b_hh,
                        const float* __restrict__ x, const float* __restrict__ h0,
                        float* __restrict__ gates) {
  __shared__ float red[256];
  int r = blockIdx.x, t = threadIdx.x;
  const float* wr; const float* vec; float bias; int oi;
  if (r < 3 * H) { wr = w_ih + (size_t)r * H;         vec = x;  bias = b_ih[r];       oi = r; }
  else           { int rr = r - 3 * H;
                   wr = w_hh + (size_t)rr * H;        vec = h0; bias = b_hh[rr];      oi = r; }
  float acc = 0.f;
  for (int i = t; i < H; i += 256) acc += wr[i] * vec[i];
  red[t] = acc; __syncthreads();
  for (int s2 = 128; s2 > 0; s2 >>= 1) { if (t < s2) red[t] += red[t + s2]; __syncthreads(); }
  if (t == 0) gates[oi] = red[0] + bias;
}

// GRU gate math -> h_new (both to d_out segment and workspace)
__global__ void k_gru(const float* __restrict__ gates, const float* __restrict__ h0,
                      float* __restrict__ hn_out, float* __restrict__ hn_ws) {
  int j = blockIdx.x * blockDim.x + threadIdx.x;
  if (j >= H) return;
  float ir = gates[j],         iz = gates[H + j],     inn = gates[2 * H + j];
  float hr = gates[3 * H + j], hz = gates[4 * H + j], hnn = gates[5 * H + j];
  float r = 1.f / (1.f + expf(-(ir + hr)));
  float z = 1.f / (1.f + expf(-(iz + hz)));
  float n = tanhf(inn + r * hnn);
  float h = (1.f - z) * n + z * h0[j];
  hn_out[j] = h;
  hn_ws[j]  = h;
}

// logits[v] = dot(out_w[v,:], h_new) + out_b[v], via V_WMMA_F32_16X16X4_F32.
// Wave owns 16 rows. Per iter: lane m loads W[row][h..h+3] (b128), lane m+16 loads
// W[row][h+4..h+7]; two WMMAs cover K = h..h+7 (A layout: lanes0-15 K=0,1; lanes16-31 K=2,3).
// B carries h_new only in column 0 (lanes 0 and 16) -> D column 0 = the 16 dot products.
__global__ void k_logits_wmma(const float* __restrict__ W, const float* __restrict__ x,
                              const float* __restrict__ bias, float* __restrict__ logits) {
  const int lane = threadIdx.x & 31;
  const int wave = blockIdx.x * (blockDim.x >> 5) + (threadIdx.x >> 5);
  const int rowBase = wave * 16;
  if (rowBase >= V) return;                 // wave-uniform exit: EXEC stays full for WMMA
  const int m = lane & 15;
  int row = rowBase + m; if (row >= V) row = V - 1;   // clamp loads; discard at store
  const int half = (lane >> 4) << 2;        // 0 or 4
  const float* wrow = W + (size_t)row * H + half;
  const float* xp   = x + half;
  const bool contrib = (m == 0);            // only column 0 of B is live
  v8f c = {};
  for (int h = 0; h < H; h += 8) {
    float4 a = *(const float4*)(wrow + h);
    float4 b = *(const float4*)(xp + h);
    float b0 = contrib ? b.x : 0.f, b1 = contrib ? b.y : 0.f;
    float b2 = contrib ? b.z : 0.f, b3 = contrib ? b.w : 0.f;
    v2f A1 = {a.x, a.y}, B1 = {b0, b1};
    v2f A2 = {a.z, a.w}, B2 = {b2, b3};
    c = __builtin_amdgcn_wmma_f32_16x16x4_f32(false, A1, false, B1, (short)0, c, false, false);
    c = __builtin_amdgcn_wmma_f32_16x16x4_f32(false, A2, false, B2, (short)0, c, false, false);
  }
  // D column 0: lane 0 -> rows rowBase+0..7 in c[0..7]; lane 16 -> rows rowBase+8..15
  if (m == 0) {
    int base = rowBase + ((lane >> 4) << 3);
    #pragma unroll
    for (int i = 0; i < 8; ++i) {
      int r = base + i;
      if (r < V) logits[r] = c[i] + bias[r];
    }
  }
}

// Single-block log_softmax over V (3 passes over ~200KB, negligible vs 206MB GEMV)
__global__ void k_logsoftmax(const float* __restrict__ logits, float* __restrict__ out) {
  __shared__ float red[256];
  int t = threadIdx.x;
  float m = -INFINITY;
  for (int i = t; i < V; i += 256) m = fmaxf(m, logits[i]);
  red[t] = m; __syncthreads();
  for (int s2 = 128; s2 > 0; s2 >>= 1) { if (t < s2) red[t] = fmaxf(red[t], red[t + s2]); __syncthreads(); }
  float gmax = red[0]; __syncthreads();
  float s = 0.f;
  for (int i = t; i < V; i += 256) s += expf(logits[i] - gmax);
  red[t] = s; __syncthreads();
  for (int s2 = 128; s2 > 0; s2 >>= 1) { if (t < s2) red[t] += red[t + s2]; __syncthreads(); }
  float lse = gmax + logf(red[0]);
  for (int i = t; i < V; i += 256) out[i] = logits[i] - lse;
}

extern "C" void kernel_launch(void* const* d_in, const int* in_sizes, int n_in,
                              void* d_out, int out_size, void* d_ws, size_t ws_size,
                              hipStream_t stream) {
  const int*   tok    = (const int*)d_in[0];
  const float* hid    = (const float*)d_in[1];   // [1,1,H]
  const float* enc    = (const float*)d_in[2];   // [S,H]
  const float* emb    = (const float*)d_in[3];   // [V,H]
  // d_in[4]=attn_w, d_in[5]=attn_b: unused — softmax of a constant is exactly uniform
  const float* comb_w = (const float*)d_in[6];   // [H,2H]
  const float* comb_b = (const float*)d_in[7];
  const float* w_ih   = (const float*)d_in[8];   // [3H,H]
  const float* w_hh   = (const float*)d_in[9];   // [3H,H]
  const float* b_ih   = (const float*)d_in[10];
  const float* b_hh   = (const float*)d_in[11];
  const float* out_w  = (const float*)d_in[12];  // [V,H]
  const float* out_b  = (const float*)d_in[13];

  float* out = (float*)d_out;   // [0,V) log-softmax | [V,V+H) h_new | [V+H,V+H+S) attn_w
  float* ws  = (float*)d_ws;
  float* logits = ws + WS_LOGITS;
  float* cat    = ws + WS_CAT;
  float* xv     = ws + WS_X;
  float* gates  = ws + WS_GATES;
  float* hn     = ws + WS_HN;
  float* part   = ws + WS_PART;

  k_fill_attn     <<<(S + 255) / 256, 256, 0, stream>>>(out + V + H);
  k_colsum_partial<<<64,              256, 0, stream>>>(enc, part);
  k_build_cat     <<<H / 256,         256, 0, stream>>>(part, tok, emb, cat);
  k_combined      <<<H,               256, 0, stream>>>(comb_w, comb_b, cat, xv);
  k_gates         <<<6 * H,           256, 0, stream>>>(w_ih, w_hh, b_ih, b_hh, xv, hid, gates);
  k_gru           <<<H / 256,         256, 0, stream>>>(gates, hid, out + V, hn);

  int nWaves  = (V + 15) / 16;
  int nBlocks = (nWaves + 7) / 8;   // 8 waves (256 threads) per block
  k_logits_wmma   <<<nBlocks,         256, 0, stream>>>(out_w, hn, out_b, logits);
  k_logsoftmax    <<<1,               256, 0, stream>>>(logits, out);
}